// Aggregator_34703335752220
// MI455X (gfx1250) — compile-verified
//
#include <hip/hip_runtime.h>
#include <hip/hip_bf16.h>
#include <cstdint>

#define N_NODES 100000
#define N_EDGES 1000000
#define D       64
#define TILE    256    // edges staged per block in the CSR-fill pass
#define SCAN_T  1024   // threads in the single-workgroup scan

// ---------------------------------------------------------------------------
// CDNA5 async global->LDS copy (ASYNCcnt-tracked DMA). One dword per lane:
//   each lane copies 4 bytes from the global address in vaddr to the LDS address in vdst
// ---------------------------------------------------------------------------
__device__ __forceinline__ void async_ld_b32_to_lds(const int* gptr, unsigned lds_byte_addr) {
    asm volatile("global_load_async_to_lds_b32 %0, %1, off"
                 :
                 : "v"(lds_byte_addr),
                   "v"((unsigned long long)(uintptr_t)gptr)
                 : "memory");
}
__device__ __forceinline__ void wait_async_zero() {
    asm volatile("s_wait_asynccnt 0" ::: "memory");
}

// ---------------------------------------------------------------------------
// Pass 0: zero both histogram arrays (out-degree for norms, in-degree bins).
// ---------------------------------------------------------------------------
__global__ void zero_kernel(unsigned* __restrict__ a, unsigned* __restrict__ b) {
    int i = blockIdx.x * blockDim.x + threadIdx.x;
    if (i < N_NODES) { a[i] = 0u; b[i] = 0u; }
}

// ---------------------------------------------------------------------------
// Pass 1: histogram both endpoints. out_deg[src]++ (norms), in_deg[dst]++
// (CSR bin sizes). 2M native u32 atomics into 800KB of L2-resident counters.
// ---------------------------------------------------------------------------
__global__ void degree_kernel(const int* __restrict__ esrc, const int* __restrict__ edst,
                              unsigned* __restrict__ out_deg, unsigned* __restrict__ in_deg) {
    int e = blockIdx.x * blockDim.x + threadIdx.x;
    if (e < N_EDGES) {
        atomicAdd(&out_deg[esrc[e]], 1u);
        atomicAdd(&in_deg[edst[e]], 1u);
    }
}

// ---------------------------------------------------------------------------
// Pass 2: single-workgroup exclusive scan of in_deg -> offsets[N+1] + cursor.
// 1024 threads x ~98-element serial chunks, Hillis-Steele over partials.
// ---------------------------------------------------------------------------
__global__ void __launch_bounds__(SCAN_T)
scan_kernel(const unsigned* __restrict__ counts,
            int* __restrict__ offsets, int* __restrict__ cursor) {
    __shared__ unsigned sh[SCAN_T];
    const int t  = threadIdx.x;
    const int CH = (N_NODES + SCAN_T - 1) / SCAN_T;   // 98
    const int b  = t * CH;
    const int e  = min(b + CH, N_NODES);

    unsigned s = 0;
    for (int i = b; i < e; ++i) s += counts[i];
    sh[t] = s;
    __syncthreads();

    for (int ofs = 1; ofs < SCAN_T; ofs <<= 1) {       // inclusive scan of partials
        unsigned v = (t >= ofs) ? sh[t - ofs] : 0u;
        __syncthreads();
        sh[t] += v;
        __syncthreads();
    }

    unsigned run = (t == 0) ? 0u : sh[t - 1];          // exclusive base for my chunk
    for (int i = b; i < e; ++i) {
        unsigned c = counts[i];
        offsets[i] = (int)run;
        cursor[i]  = (int)run;
        run += c;
    }
    if (b < N_NODES && e == N_NODES) offsets[N_NODES] = (int)run;  // total = N_EDGES
}

// ---------------------------------------------------------------------------
// Pass 3: out_deg -> 1/sqrt(out_deg) in place (every node has out_deg >= 1).
// ---------------------------------------------------------------------------
__global__ void rsqrt_kernel(float* __restrict__ deg) {
    int i = blockIdx.x * blockDim.x + threadIdx.x;
    if (i < N_NODES) {
        unsigned d = reinterpret_cast<unsigned*>(deg)[i];
        deg[i] = 1.0f / sqrtf((float)d);
    }
}

// ---------------------------------------------------------------------------
// Pass 4: counting-sort fill of CSR. Edge (src,dst) pairs staged into LDS via
// async DMA (each lane's slot is lane-private -> no barrier needed), then one
// u32 atomic per edge reserves the slot and src is scattered into csr_src.
// ---------------------------------------------------------------------------
__global__ void __launch_bounds__(TILE)
csr_fill_kernel(const int* __restrict__ esrc, const int* __restrict__ edst,
                int* __restrict__ cursor, int* __restrict__ csr_src) {
    __shared__ int lds_src[TILE];
    __shared__ int lds_dst[TILE];

    const int tid  = threadIdx.x;
    const int base = blockIdx.x * TILE;
    const int e    = base + tid;

    if (e < N_EDGES) {
        async_ld_b32_to_lds(esrc + e, (unsigned)(uintptr_t)&lds_src[tid]);
        async_ld_b32_to_lds(edst + e, (unsigned)(uintptr_t)&lds_dst[tid]);
    }
    wait_async_zero();

    if (e < N_EDGES) {
        const int s = lds_src[tid];
        const int t = lds_dst[tid];
        const int pos = atomicAdd(&cursor[t], 1);
        csr_src[pos] = s;
    }
}

// ---------------------------------------------------------------------------
// Pass 5: atomic-free gather. One wave per node; 32 lanes x float2 = the full
// 64-float row. Per 32-edge chunk: coalesced csr_src load + sqrt_deg gather,
// wave-wide prefetch of all 32 source rows (both 128B halves), then
// shfl-broadcast each (src, scale) and FMA into register accumulators.
// One coalesced 256B store per node.
// ---------------------------------------------------------------------------
__global__ void __launch_bounds__(256)
gather_kernel(const float* __restrict__ embed,
              const int*   __restrict__ csr_src,
              const int*   __restrict__ offsets,
              const float* __restrict__ sqrt_deg,
              float*       __restrict__ out) {
    const int lane = threadIdx.x & 31;
    const int node = blockIdx.x * 8 + (threadIdx.x >> 5);
    if (node >= N_NODES) return;

    const int beg = offsets[node];
    const int end = offsets[node + 1];

    float ax = 0.f, ay = 0.f;
    for (int chunk = beg; chunk < end; chunk += 32) {
        const int k = chunk + lane;
        int   sv  = 0;
        float scv = 0.f;
        if (k < end) {
            sv  = csr_src[k];                               // coalesced b32
            scv = sqrt_deg[sv];                             // gathered dword
            // warm the full 256B row this lane will broadcast (2 x 128B lines)
            __builtin_prefetch(&embed[(size_t)sv * D], 0, 0);
            __builtin_prefetch(&embed[(size_t)sv * D + 32], 0, 0);
        }
        const int m = min(32, end - chunk);                 // wave-uniform
        for (int j = 0; j < m; ++j) {
            const int   s  = __shfl(sv,  j, 32);            // wave32 broadcast
            const float sc = __shfl(scv, j, 32);
            const float2 v = *reinterpret_cast<const float2*>(
                &embed[(size_t)s * D + lane * 2]);          // 256B/wave coalesced
            ax = fmaf(v.x, sc, ax);
            ay = fmaf(v.y, sc, ay);
        }
    }

    const float scn = sqrt_deg[node];
    float2 r;
    r.x = ax * scn;
    r.y = ay * scn;
    *reinterpret_cast<float2*>(&out[(size_t)node * D + lane * 2]) = r;
}

// ---------------------------------------------------------------------------
// Workspace layout (byte offsets, 128B-aligned):
//   sqrt_deg (u32 out-deg -> f32 norm) : [0,        400128)
//   in_deg   (u32)                     : [400128,   800256)
//   offsets  (int[N+1])                : [800256,  1200512)
//   cursor   (int[N])                  : [1200512, 1600640)
//   csr_src  (int[N_EDGES])            : [1600640, 5600640)   ~5.6 MB total
// ---------------------------------------------------------------------------
extern "C" void kernel_launch(void* const* d_in, const int* in_sizes, int n_in,
                              void* d_out, int out_size, void* d_ws, size_t ws_size,
                              hipStream_t stream) {
    const float* embed = (const float*)d_in[0];
    const int*   esrc  = (const int*)d_in[1];
    const int*   edst  = (const int*)d_in[2];

    uint8_t* ws = (uint8_t*)d_ws;
    float*    sqrt_deg = (float*)   (ws + 0);
    unsigned* in_deg   = (unsigned*)(ws + 400128);
    int*      offsets  = (int*)     (ws + 800256);
    int*      cursor   = (int*)     (ws + 1200512);
    int*      csr_src  = (int*)     (ws + 1600640);
    float*    out      = (float*)d_out;

    zero_kernel<<<(N_NODES + 255) / 256, 256, 0, stream>>>((unsigned*)sqrt_deg, in_deg);

    degree_kernel<<<(N_EDGES + 255) / 256, 256, 0, stream>>>(
        esrc, edst, (unsigned*)sqrt_deg, in_deg);

    scan_kernel<<<1, SCAN_T, 0, stream>>>(in_deg, offsets, cursor);

    rsqrt_kernel<<<(N_NODES + 255) / 256, 256, 0, stream>>>(sqrt_deg);

    csr_fill_kernel<<<(N_EDGES + TILE - 1) / TILE, TILE, 0, stream>>>(
        esrc, edst, cursor, csr_src);

    gather_kernel<<<(N_NODES + 7) / 8, 256, 0, stream>>>(
        embed, csr_src, offsets, sqrt_deg, out);
}